// compute_TE_45801531244676
// MI455X (gfx1250) — compile-verified
//
#include <hip/hip_runtime.h>
#include <math.h>

typedef __attribute__((ext_vector_type(2))) float v2f;
typedef __attribute__((ext_vector_type(8))) float v8f;

#define NN 4096
#define TT 50
#define EMBD 64
#define GC 256            // 4*H gate columns
#define LSTM_ROWS 64      // batch rows per workgroup
#define LSTM_THREADS 512  // 16 waves = 4 M-tiles x 4 hidden-tiles

__device__ __forceinline__ float sigmoidf_(float x) { return 1.0f / (1.0f + __expf(-x)); }

// ---------------- K0: fold linear layers into small tables ----------------
// Acomb = W_ih @ W_speed (256x2); biasc = W_ih@b_speed + b_ih + b_hh (256)
// wf/wg = conv-ch0/ch1 folded with lin_w (64); Cc = map term + lin_b (1)
__global__ __launch_bounds__(256) void k0_setup(
    const float* __restrict__ W_speed, const float* __restrict__ b_speed,
    const float* __restrict__ W_ih, const float* __restrict__ b_ih,
    const float* __restrict__ b_hh, const float* __restrict__ conv_w,
    const float* __restrict__ lin_w, const float* __restrict__ lin_b,
    const float* __restrict__ map_e,
    float* __restrict__ Acomb, float* __restrict__ biasc,
    float* __restrict__ wf, float* __restrict__ wg, float* __restrict__ Cc)
{
  int t = threadIdx.x;
  if (t < GC) {
    float a0 = 0.f, a1 = 0.f, bc = 0.f;
    for (int e = 0; e < EMBD; e++) {
      float w = W_ih[t * EMBD + e];
      a0 += w * W_speed[e * 2 + 0];
      a1 += w * W_speed[e * 2 + 1];
      bc += w * b_speed[e];
    }
    Acomb[t * 2 + 0] = a0; Acomb[t * 2 + 1] = a1;
    biasc[t] = bc + b_ih[t] + b_hh[t];
  }
  __shared__ float sm[EMBD];
  if (t < EMBD) {
    float v  = lin_w[t];
    float vp = (t + 1 < EMBD) ? lin_w[t + 1] : 0.0f;
    float vm = (t > 0)        ? lin_w[t - 1] : 0.0f;
    // w[e] = k0*v[e+1] + k1*v[e] + k2*v[e-1]  (conv then dot-with-v, transposed)
    wf[t] = conv_w[0] * vp + conv_w[1] * v + conv_w[2] * vm;
    wg[t] = conv_w[3] * vp + conv_w[4] * v + conv_w[5] * vm;
    float wm = conv_w[6] * vp + conv_w[7] * v + conv_w[8] * vm;
    sm[t] = map_e[t] * wm;
  }
  __syncthreads();
  if (t == 0) { float s = 0.f; for (int e = 0; e < EMBD; e++) s += sm[e]; Cc[0] = s + lin_b[0]; }
}

// ---------------- K1: LSTM, recurrence fully on-chip, fp32 WMMA ----------------
// Wave (mtile, ht) owns the 16(batch)x16(hidden) state tile. Its W_hh fragments
// (4 gates x 16 k-steps) live in 128 VGPRs for the entire 50-step recurrence;
// only the A operand (previous h) and the h-update touch LDS each step.
__global__ __launch_bounds__(LSTM_THREADS) void k1_lstm(
    const float* __restrict__ speed, const float* __restrict__ W_hh,
    const float* __restrict__ Acomb, const float* __restrict__ biasc,
    float* __restrict__ node)
{
  __shared__ float sH[LSTM_ROWS * 65];   // [localRow][k], stride 65 = bank-conflict-free
  __shared__ float sSp[LSTM_ROWS * 2];
  __shared__ float sAc[GC * 2];
  __shared__ float sBc[GC];

  const int tid = threadIdx.x;
  const int blockRow0 = blockIdx.x * LSTM_ROWS;

  for (int idx = tid; idx < LSTM_ROWS * 65; idx += LSTM_THREADS) sH[idx] = 0.0f;
  for (int idx = tid; idx < GC * 2; idx += LSTM_THREADS) sAc[idx] = Acomb[idx];
  for (int idx = tid; idx < GC; idx += LSTM_THREADS) sBc[idx] = biasc[idx];

  const int wave = tid >> 5, lane = tid & 31;
  const int mtile = wave >> 2;           // which 16 batch rows
  const int ht    = wave & 3;            // which 16 hidden columns
  const int half = lane >> 4, lcol = lane & 15;
  const int arow = mtile * 16 + lcol;    // A-fragment row for this lane

  // Preload register-resident B fragments: barr[kk*4+q] = W_hh^T tile piece.
  // B layout per ISA: lane<16 -> (K=4kk{+1}, N=lane); lane>=16 -> (K=4kk+2{+3}, N=lane-16)
  v2f barr[64];
#pragma unroll
  for (int kk = 0; kk < 16; kk++) {
    int k = 4 * kk + 2 * half;
#pragma unroll
    for (int q = 0; q < 4; q++) {
      int col = q * 64 + ht * 16 + lcol; // gate column in [0,256)
      const float* p = W_hh + (size_t)col * EMBD + k;
      barr[kk * 4 + q].x = p[0];
      barr[kk * 4 + q].y = p[1];
    }
  }

  float creg[8];                         // cell state: 8 rows of the 16x16 tile per lane
  float hnew[8];
#pragma unroll
  for (int i = 0; i < 8; i++) { creg[i] = 0.0f; hnew[i] = 0.0f; }
  __syncthreads();

  for (int tau = 0; tau < TT; tau++) {
    if (tid < LSTM_ROWS) { // stage this step's (scrambled-reshape) speed rows
      int row = blockRow0 + tid;
      sSp[tid * 2 + 0] = speed[(size_t)(tau * NN + row) * 2 + 0];
      sSp[tid * 2 + 1] = speed[(size_t)(tau * NN + row) * 2 + 1];
    }
    __syncthreads();

    v8f acc[4];                          // i, f, g, o accumulators (16x16 each)
#pragma unroll
    for (int r = 0; r < 8; r++) {        // init C with input projection + bias
      int M = r + 8 * half;
      int lr = mtile * 16 + M;
      float sx = sSp[lr * 2 + 0], sy = sSp[lr * 2 + 1];
#pragma unroll
      for (int q = 0; q < 4; q++) {
        int col = q * 64 + ht * 16 + lcol;
        acc[q][r] = sBc[col] + sx * sAc[col * 2 + 0] + sy * sAc[col * 2 + 1];
      }
    }
#pragma unroll
    for (int kk = 0; kk < 16; kk++) {    // K = 64 in steps of 4; A shared by 4 gates
      int k = 4 * kk + 2 * half;
      v2f a; a.x = sH[arow * 65 + k]; a.y = sH[arow * 65 + k + 1];
#pragma unroll
      for (int q = 0; q < 4; q++) {
        acc[q] = __builtin_amdgcn_wmma_f32_16x16x4_f32(
            false, a, false, barr[kk * 4 + q], (short)0, acc[q], false, false);
      }
    }
#pragma unroll
    for (int r = 0; r < 8; r++) {        // gate nonlinearity + state update
      float iv = sigmoidf_(acc[0][r]);
      float fv = sigmoidf_(acc[1][r]);
      float gg = tanhf(acc[2][r]);
      float ov = sigmoidf_(acc[3][r]);
      float c  = fv * creg[r] + iv * gg;
      creg[r] = c;
      hnew[r] = ov * tanhf(c);
    }
    __syncthreads();                     // all waves done reading old sH
#pragma unroll
    for (int r = 0; r < 8; r++) {
      int M = r + 8 * half;
      sH[(mtile * 16 + M) * 65 + ht * 16 + lcol] = hnew[r];
    }
    __syncthreads();                     // new sH visible before next step's reads
  }

#pragma unroll
  for (int r = 0; r < 8; r++) {
    int M = r + 8 * half;
    node[(size_t)(blockRow0 + mtile * 16 + M) * EMBD + ht * 16 + lcol] = hnew[r];
  }
}

// ---------------- K2: f_i = node_i . wf ; g_i = node_i . wg ----------------
__global__ __launch_bounds__(256) void k2_fg(
    const float* __restrict__ node, const float* __restrict__ wf,
    const float* __restrict__ wg, float* __restrict__ fv, float* __restrict__ gv)
{
  int i = blockIdx.x * 256 + threadIdx.x;
  if (i < NN) {
    float sf = 0.f, sg = 0.f;
    const float* row = node + (size_t)i * EMBD;
#pragma unroll
    for (int e = 0; e < EMBD; e++) { float x = row[e]; sf += x * wf[e]; sg += x * wg[e]; }
    fv[i] = sf; gv[i] = sg;
  }
}

// ---------------- K3: per-row stats of TE (one streaming pass) ----------------
__global__ __launch_bounds__(256) void k3_row(
    const float* __restrict__ TE, float* __restrict__ R, int* __restrict__ rowArg,
    float* __restrict__ rowSq, float* __restrict__ rowMax)
{
  __shared__ float sS[256], sQ[256], sM[256];
  __shared__ int sI[256];
  int row = blockIdx.x, tid = threadIdx.x;
  const float* p = TE + (size_t)row * NN;
  float s = 0.f, q = 0.f, mx = -INFINITY; int mi = NN;
  for (int j = tid; j < NN; j += 256) {
    float v = p[j];
    q += v * v;                 // full sum of squares (incl. diag)
    if (j != row) s += v;       // row sum excluding diagonal
    if (v > mx) { mx = v; mi = j; }
  }
  sS[tid] = s; sQ[tid] = q; sM[tid] = mx; sI[tid] = mi;
  __syncthreads();
  for (int st = 128; st > 0; st >>= 1) {
    if (tid < st) {
      sS[tid] += sS[tid + st]; sQ[tid] += sQ[tid + st];
      float om = sM[tid + st]; int oi = sI[tid + st];
      if (om > sM[tid] || (om == sM[tid] && oi < sI[tid])) { sM[tid] = om; sI[tid] = oi; }
    }
    __syncthreads();
  }
  if (tid == 0) { R[row] = sS[0]; rowSq[row] = sQ[0]; rowMax[row] = sM[0]; rowArg[row] = sI[0]; }
}

// ---------------- K4: per-column partials over 64-row strips (L2-resident pass) ---
__global__ __launch_bounds__(256) void k4_col(
    const float* __restrict__ TE, float* __restrict__ csum,
    float* __restrict__ cmax, int* __restrict__ carg)
{
  int c = blockIdx.x * 256 + threadIdx.x;
  int strip = blockIdx.y;
  int r0 = strip * 64;
  float s = 0.f, mx = -INFINITY; int mi = NN;
  for (int r = r0; r < r0 + 64; r++) {
    float v = TE[(size_t)r * NN + c];
    if (r != c) s += v;
    if (v > mx) { mx = v; mi = r; }
  }
  size_t o = (size_t)strip * NN + c;
  csum[o] = s; cmax[o] = mx; carg[o] = mi;
}

// ---------------- K5: deterministic finalize (loss + acc) ----------------
__global__ __launch_bounds__(256) void k5_final(
    const float* __restrict__ fv, const float* __restrict__ gv,
    const float* __restrict__ R, const int* __restrict__ rowArg,
    const float* __restrict__ rowSq, const float* __restrict__ rowMax,
    const float* __restrict__ csum, const float* __restrict__ cmax,
    const int* __restrict__ carg, const float* __restrict__ Cc,
    float* __restrict__ colS, int* __restrict__ colArgF, float* __restrict__ out)
{
  __shared__ double sd[256];
  __shared__ float sfl[256];
  __shared__ int si[256];
  __shared__ int shJ1, shJ2, shI1, shI2;
  __shared__ float shG1, shG2, shF1, shF2;
  int tid = threadIdx.x;
  float C = Cc[0];

  // A: combine column partials in ascending-strip order (first-index ties)
  for (int j = tid; j < NN; j += 256) {
    double s = 0.0; float mx = -INFINITY; int mi = NN;
    for (int st = 0; st < 64; st++) {
      size_t o = (size_t)st * NN + j;
      s += (double)csum[o];
      float m = cmax[o];
      if (m > mx) { mx = m; mi = carg[o]; }
    }
    colS[j] = (float)s; colArgF[j] = mi;
  }
  __syncthreads();

  // B: scalar sums (double accumulators, fixed-order tree -> deterministic)
  double Sf = 0, Sg = 0, Sf2 = 0, Sg2 = 0, Sdiag = 0, SfR = 0, SgC = 0, Soff = 0, Ssq = 0;
  float mT = -INFINITY;
  for (int i = tid; i < NN; i += 256) {
    double fi = fv[i], gi = gv[i], Ri = R[i];
    Sf += fi; Sg += gi; Sf2 += fi * fi; Sg2 += gi * gi;
    double d = fi + gi + (double)C; Sdiag += d * d;
    SfR += fi * Ri; SgC += gi * (double)colS[i]; Soff += Ri; Ssq += (double)rowSq[i];
    float m = rowMax[i]; if (m > mT) mT = m;
  }
  auto redD = [&](double v) -> double {
    sd[tid] = v; __syncthreads();
    for (int st = 128; st > 0; st >>= 1) { if (tid < st) sd[tid] += sd[tid + st]; __syncthreads(); }
    double r = sd[0]; __syncthreads(); return r;
  };
  Sf = redD(Sf); Sg = redD(Sg); Sf2 = redD(Sf2); Sg2 = redD(Sg2); Sdiag = redD(Sdiag);
  SfR = redD(SfR); SgC = redD(SgC); Soff = redD(Soff); Ssq = redD(Ssq);
  sfl[tid] = mT; __syncthreads();
  for (int st = 128; st > 0; st >>= 1) {
    if (tid < st && sfl[tid + st] > sfl[tid]) sfl[tid] = sfl[tid + st];
    __syncthreads();
  }
  mT = sfl[0]; __syncthreads();

  // C: top-2 of g and f with first-index semantics (serial; O(N))
  if (tid == 0) {
    int j1 = 0; float g1 = gv[0];
    for (int j = 1; j < NN; j++) { float v = gv[j]; if (v > g1) { g1 = v; j1 = j; } }
    int j2 = -1; float g2 = -INFINITY;
    for (int j = 0; j < NN; j++) { if (j == j1) continue; float v = gv[j]; if (v > g2) { g2 = v; j2 = j; } }
    int i1 = 0; float f1 = fv[0];
    for (int i = 1; i < NN; i++) { float v = fv[i]; if (v > f1) { f1 = v; i1 = i; } }
    int i2 = -1; float f2 = -INFINITY;
    for (int i = 0; i < NN; i++) { if (i == i1) continue; float v = fv[i]; if (v > f2) { f2 = v; i2 = i; } }
    shJ1 = j1; shJ2 = j2; shG1 = g1; shG2 = g2;
    shI1 = i1; shI2 = i2; shF1 = f1; shF2 = f2;
  }
  __syncthreads();

  // D: accuracy counts; pred(i,j) = (i==j) ? 0 : f_i+g_j+C, argmax = first max
  int cnt = 0;
  for (int i = tid; i < NN; i += 256) {
    float gb = (i == shJ1) ? shG2 : shG1;
    int   js = (i == shJ1) ? shJ2 : shJ1;
    float pv = fv[i] + gb + C;
    int pa = (pv > 0.0f) ? js : ((pv < 0.0f) ? i : (i < js ? i : js));
    if (pa == rowArg[i]) cnt++;
  }
  for (int j = tid; j < NN; j += 256) {
    float fb = (j == shI1) ? shF2 : shF1;
    int   is = (j == shI1) ? shI2 : shI1;
    float pv = fb + gv[j] + C;
    int pa = (pv > 0.0f) ? is : ((pv < 0.0f) ? j : (j < is ? j : is));
    if (pa == colArgF[j]) cnt++;
  }
  si[tid] = cnt; __syncthreads();
  for (int st = 128; st > 0; st >>= 1) { if (tid < st) si[tid] += si[tid + st]; __syncthreads(); }

  if (tid == 0) {
    double n = (double)NN;
    double s = (mT > 1e-3f) ? (1.0 / (double)mT) : 1.0;
    double Cd = (double)C;
    // sum over i!=j of (f_i+g_j+C)^2, closed form minus diagonal terms
    double SP2 = n * Sf2 + n * Sg2 + n * n * Cd * Cd + 2.0 * Sf * Sg
               + 2.0 * Cd * n * Sf + 2.0 * Cd * n * Sg - Sdiag;
    double cross = SfR + SgC + Cd * Soff;       // sum_{i!=j} pred * TE
    double loss = (SP2 - 2.0 * s * cross + s * s * Ssq) / (n * n);
    double acc = (double)si[0] / (2.0 * n);
    out[0] = (float)loss;
    out[1] = (float)acc;
  }
}

extern "C" void kernel_launch(void* const* d_in, const int* in_sizes, int n_in,
                              void* d_out, int out_size, void* d_ws, size_t ws_size,
                              hipStream_t stream)
{
  const float* TE    = (const float*)d_in[0];
  // d_in[1] = xy (unused by the reference)
  const float* speed = (const float*)d_in[2];
  const float* map_e = (const float*)d_in[3];
  const float* W_sp  = (const float*)d_in[4];
  const float* b_sp  = (const float*)d_in[5];
  const float* W_ih  = (const float*)d_in[6];
  const float* W_hh  = (const float*)d_in[7];
  const float* b_ih  = (const float*)d_in[8];
  const float* b_hh  = (const float*)d_in[9];
  const float* convw = (const float*)d_in[10];
  const float* linw  = (const float*)d_in[11];
  const float* linb  = (const float*)d_in[12];
  float* out = (float*)d_out;

  char* w = (char*)d_ws;
  auto alloc = [&](size_t bytes) -> char* {
    char* p = w; w += (bytes + 255) & ~(size_t)255; return p;
  };
  float* node    = (float*)alloc((size_t)NN * EMBD * 4);
  float* fv      = (float*)alloc((size_t)NN * 4);
  float* gv      = (float*)alloc((size_t)NN * 4);
  float* Acomb   = (float*)alloc((size_t)GC * 2 * 4);
  float* biasc   = (float*)alloc((size_t)GC * 4);
  float* wf      = (float*)alloc((size_t)EMBD * 4);
  float* wg      = (float*)alloc((size_t)EMBD * 4);
  float* Cc      = (float*)alloc(256);
  float* R       = (float*)alloc((size_t)NN * 4);
  int*   rowArg  = (int*)alloc((size_t)NN * 4);
  float* rowSq   = (float*)alloc((size_t)NN * 4);
  float* rowMax  = (float*)alloc((size_t)NN * 4);
  float* csum    = (float*)alloc((size_t)64 * NN * 4);
  float* cmax    = (float*)alloc((size_t)64 * NN * 4);
  int*   carg    = (int*)alloc((size_t)64 * NN * 4);
  float* colS    = (float*)alloc((size_t)NN * 4);
  int*   colArgF = (int*)alloc((size_t)NN * 4);

  hipLaunchKernelGGL(k0_setup, dim3(1), dim3(256), 0, stream,
                     W_sp, b_sp, W_ih, b_ih, b_hh, convw, linw, linb, map_e,
                     Acomb, biasc, wf, wg, Cc);
  hipLaunchKernelGGL(k1_lstm, dim3(NN / LSTM_ROWS), dim3(LSTM_THREADS), 0, stream,
                     speed, W_hh, Acomb, biasc, node);
  hipLaunchKernelGGL(k2_fg, dim3(NN / 256), dim3(256), 0, stream, node, wf, wg, fv, gv);
  hipLaunchKernelGGL(k3_row, dim3(NN), dim3(256), 0, stream, TE, R, rowArg, rowSq, rowMax);
  hipLaunchKernelGGL(k4_col, dim3(NN / 256, 64), dim3(256), 0, stream, TE, csum, cmax, carg);
  hipLaunchKernelGGL(k5_final, dim3(1), dim3(256), 0, stream,
                     fv, gv, R, rowArg, rowSq, rowMax, csum, cmax, carg, Cc,
                     colS, colArgF, out);
}